// GraphNet_58686433132830
// MI455X (gfx1250) — compile-verified
//
#include <hip/hip_runtime.h>

typedef __attribute__((ext_vector_type(2))) float v2f;
typedef __attribute__((ext_vector_type(8))) float v8f;

#define NNODES 50000
#define NEDGES 800000
#define DIM    64
#define DIM2   128
#define EPS_MSG 1e-7f
#define BN_EPS  1e-5f

// ---------------- Edge pass 1: msg = relu(x[src]+e)+eps ; segment max via uint atomicMax
__global__ void genconv_edge_max(const float* __restrict__ x,
                                 const long long* __restrict__ src,
                                 const long long* __restrict__ dst,
                                 const float* __restrict__ ea,
                                 unsigned int* __restrict__ mx)
{
    int t = blockIdx.x * blockDim.x + threadIdx.x;
    int e = t >> 4;                 // 16 threads per edge (4 channels each)
    int c = (t & 15) << 2;
    if (e >= NEDGES) return;
    int s = (int)src[e];
    int d = (int)dst[e];
    float4 xv = *(const float4*)(x + (size_t)s * DIM + c);
    float4 ev = *(const float4*)(ea + (size_t)e * DIM + c);
    float m0 = fmaxf(xv.x + ev.x, 0.f) + EPS_MSG;
    float m1 = fmaxf(xv.y + ev.y, 0.f) + EPS_MSG;
    float m2 = fmaxf(xv.z + ev.z, 0.f) + EPS_MSG;
    float m3 = fmaxf(xv.w + ev.w, 0.f) + EPS_MSG;
    unsigned int* p = mx + (size_t)d * DIM + c;   // all msgs > 0 -> uint order == float order
    atomicMax(p + 0, __float_as_uint(m0));
    atomicMax(p + 1, __float_as_uint(m1));
    atomicMax(p + 2, __float_as_uint(m2));
    atomicMax(p + 3, __float_as_uint(m3));
}

// ---------------- Edge pass 2: recompute msg, accumulate exp-sum and msg*exp sum
__global__ void genconv_edge_sum(const float* __restrict__ x,
                                 const long long* __restrict__ src,
                                 const long long* __restrict__ dst,
                                 const float* __restrict__ ea,
                                 const float* __restrict__ mx,
                                 float* __restrict__ ssum,
                                 float* __restrict__ wsum)
{
    int t = blockIdx.x * blockDim.x + threadIdx.x;
    int e = t >> 4;
    int c = (t & 15) << 2;
    if (e >= NEDGES) return;
    int s = (int)src[e];
    int d = (int)dst[e];
    float4 xv = *(const float4*)(x + (size_t)s * DIM + c);
    float4 ev = *(const float4*)(ea + (size_t)e * DIM + c);
    float4 mv = *(const float4*)(mx + (size_t)d * DIM + c);
    float m0 = fmaxf(xv.x + ev.x, 0.f) + EPS_MSG;
    float m1 = fmaxf(xv.y + ev.y, 0.f) + EPS_MSG;
    float m2 = fmaxf(xv.z + ev.z, 0.f) + EPS_MSG;
    float m3 = fmaxf(xv.w + ev.w, 0.f) + EPS_MSG;
    float e0 = expf(m0 - mv.x);
    float e1 = expf(m1 - mv.y);
    float e2 = expf(m2 - mv.z);
    float e3 = expf(m3 - mv.w);
    size_t base = (size_t)d * DIM + c;
    atomicAdd(ssum + base + 0, e0);
    atomicAdd(ssum + base + 1, e1);
    atomicAdd(ssum + base + 2, e2);
    atomicAdd(ssum + base + 3, e3);
    atomicAdd(wsum + base + 0, m0 * e0);
    atomicAdd(wsum + base + 1, m1 * e1);
    atomicAdd(wsum + base + 2, m2 * e2);
    atomicAdd(wsum + base + 3, m3 * e3);
}

// ---------------- h = wsum / max(ssum,1e-16) + x   (softmax-agg + residual)
__global__ void genconv_combine(const float* __restrict__ xin,
                                const float* __restrict__ ssum,
                                const float* __restrict__ wsum,
                                float* __restrict__ h)
{
    int t = blockIdx.x * blockDim.x + threadIdx.x;   // over N*DIM/4
    float4 sv = ((const float4*)ssum)[t];
    float4 wv = ((const float4*)wsum)[t];
    float4 xv = ((const float4*)xin)[t];
    float4 o;
    o.x = wv.x / fmaxf(sv.x, 1e-16f) + xv.x;
    o.y = wv.y / fmaxf(sv.y, 1e-16f) + xv.y;
    o.z = wv.z / fmaxf(sv.z, 1e-16f) + xv.z;
    o.w = wv.w / fmaxf(sv.w, 1e-16f) + xv.w;
    ((float4*)h)[t] = o;
}

// ---------------- GEMM1: h1 = h @ W1 + b1, plus BN partial sums.
// One wave per 16x16 output tile; V_WMMA_F32_16X16X4_F32, K=64 -> 16 steps.
__global__ void genconv_gemm1(const float* __restrict__ h,
                              const float* __restrict__ W1,
                              const float* __restrict__ b1,
                              float* __restrict__ h1,
                              float* __restrict__ bnsum,
                              float* __restrict__ bnsq)
{
    const int lane = threadIdx.x & 31;
    const int coltile = threadIdx.x >> 5;   // 8 waves = 8 col tiles of 128 cols
    const int rowtile = blockIdx.x;         // 3125 row tiles
    const int m  = lane & 15;
    const int hf = lane >> 4;               // wave half: K pair selector
    const int row = rowtile * 16 + m;
    const int col = coltile * 16 + m;

    v8f acc = {0.f, 0.f, 0.f, 0.f, 0.f, 0.f, 0.f, 0.f};
#pragma unroll
    for (int k0 = 0; k0 < DIM; k0 += 4) {
        int k = k0 + 2 * hf;
        float2 av = *(const float2*)(h + (size_t)row * DIM + k);
        v2f a = {av.x, av.y};
        v2f b = {W1[(size_t)k * DIM2 + col], W1[(size_t)(k + 1) * DIM2 + col]};
        acc = __builtin_amdgcn_wmma_f32_16x16x4_f32(
            false, a, false, b, (short)0, acc, false, false);
    }
    float bias = b1[col];
    float s = 0.f, sq = 0.f;
#pragma unroll
    for (int r = 0; r < 8; ++r) {
        int rr = rowtile * 16 + r + 8 * hf;   // C/D layout: row = r + 8*half
        float v = acc[r] + bias;
        h1[(size_t)rr * DIM2 + col] = v;
        s  += v;
        sq += v * v;
    }
    s  += __shfl_xor(s, 16, 32);
    sq += __shfl_xor(sq, 16, 32);
    if (hf == 0) {
        atomicAdd(bnsum + col, s);
        atomicAdd(bnsq  + col, sq);
    }
}

// ---------------- BN finalize: per-channel scale/shift
__global__ void genconv_bnfinal(const float* __restrict__ bnsum,
                                const float* __restrict__ bnsq,
                                const float* __restrict__ gamma,
                                const float* __restrict__ beta,
                                float* __restrict__ scale,
                                float* __restrict__ shift)
{
    int c = threadIdx.x;            // 128 threads
    float mu  = bnsum[c] * (1.0f / (float)NNODES);
    float var = bnsq[c]  * (1.0f / (float)NNODES) - mu * mu;
    float inv = rsqrtf(var + BN_EPS);
    float sc = gamma[c] * inv;
    scale[c] = sc;
    shift[c] = beta[c] - mu * sc;
}

// ---------------- GEMM2: xout = relu(h1*scale+shift) @ W2 + b2
__global__ void genconv_gemm2(const float* __restrict__ h1,
                              const float* __restrict__ W2,
                              const float* __restrict__ b2,
                              const float* __restrict__ scale,
                              const float* __restrict__ shift,
                              float* __restrict__ xout)
{
    __shared__ float s_sc[DIM2];
    __shared__ float s_sh[DIM2];
    if (threadIdx.x < DIM2) {
        s_sc[threadIdx.x] = scale[threadIdx.x];
        s_sh[threadIdx.x] = shift[threadIdx.x];
    }
    __syncthreads();

    const int lane = threadIdx.x & 31;
    const int coltile = threadIdx.x >> 5;   // 4 waves = 4 col tiles of 64 cols
    const int rowtile = blockIdx.x;
    const int m  = lane & 15;
    const int hf = lane >> 4;
    const int row = rowtile * 16 + m;
    const int col = coltile * 16 + m;

    v8f acc = {0.f, 0.f, 0.f, 0.f, 0.f, 0.f, 0.f, 0.f};
#pragma unroll
    for (int k0 = 0; k0 < DIM2; k0 += 4) {
        int k = k0 + 2 * hf;
        float2 av = *(const float2*)(h1 + (size_t)row * DIM2 + k);
        float a0 = fmaxf(av.x * s_sc[k]     + s_sh[k],     0.f);
        float a1 = fmaxf(av.y * s_sc[k + 1] + s_sh[k + 1], 0.f);
        v2f a = {a0, a1};
        v2f b = {W2[(size_t)k * DIM + col], W2[(size_t)(k + 1) * DIM + col]};
        acc = __builtin_amdgcn_wmma_f32_16x16x4_f32(
            false, a, false, b, (short)0, acc, false, false);
    }
    float bias = b2[col];
#pragma unroll
    for (int r = 0; r < 8; ++r) {
        int rr = rowtile * 16 + r + 8 * hf;
        xout[(size_t)rr * DIM + col] = acc[r] + bias;
    }
}

extern "C" void kernel_launch(void* const* d_in, const int* in_sizes, int n_in,
                              void* d_out, int out_size, void* d_ws, size_t ws_size,
                              hipStream_t stream)
{
    const float*     x0    = (const float*)d_in[0];
    const long long* ei    = (const long long*)d_in[1];
    const float*     ea    = (const float*)d_in[2];
    const float*     W1    = (const float*)d_in[3];
    const float*     b1    = (const float*)d_in[4];
    const float*     gamma = (const float*)d_in[5];
    const float*     beta  = (const float*)d_in[6];
    const float*     W2    = (const float*)d_in[7];
    const float*     b2    = (const float*)d_in[8];
    float* out = (float*)d_out;

    const size_t nd = (size_t)NNODES * DIM;
    float* ws    = (float*)d_ws;
    float* mx    = ws;                        // nd   (also used as uint)
    float* ssum  = mx    + nd;                // nd
    float* wsum  = ssum  + nd;                // nd
    float* bnsum = wsum  + nd;                // 128
    float* bnsq  = bnsum + DIM2;              // 128
    float* h     = bnsq  + DIM2;              // nd
    float* h1    = h     + nd;                // N*128
    float* scale = h1 + (size_t)NNODES * DIM2;// 128
    float* shift = scale + DIM2;              // 128
    float* xb0   = shift + DIM2;              // nd
    float* xb1   = xb0   + nd;                // nd

    const long long* srcp = ei;
    const long long* dstp = ei + NEDGES;

    const int edge_blocks = (NEDGES * 16) / 256;   // 50000
    const int node_blocks = (NNODES * DIM / 4) / 256; // 3125
    const int tile_blocks = NNODES / 16;           // 3125

    const float* xin = x0;
    for (int layer = 0; layer < 3; ++layer) {
        float* xout = (layer == 0) ? xb0 : (layer == 1) ? xb1 : out;

        // zero mx | ssum | wsum | bnsum | bnsq in one contiguous memset
        hipMemsetAsync(mx, 0, (3 * nd + 2 * DIM2) * sizeof(float), stream);

        genconv_edge_max<<<edge_blocks, 256, 0, stream>>>(
            xin, srcp, dstp, ea, (unsigned int*)mx);
        genconv_edge_sum<<<edge_blocks, 256, 0, stream>>>(
            xin, srcp, dstp, ea, mx, ssum, wsum);
        genconv_combine<<<node_blocks, 256, 0, stream>>>(xin, ssum, wsum, h);
        genconv_gemm1<<<tile_blocks, 256, 0, stream>>>(
            h, W1 + (size_t)layer * DIM * DIM2, b1 + (size_t)layer * DIM2,
            h1, bnsum, bnsq);
        genconv_bnfinal<<<1, DIM2, 0, stream>>>(
            bnsum, bnsq, gamma + (size_t)layer * DIM2, beta + (size_t)layer * DIM2,
            scale, shift);
        genconv_gemm2<<<tile_blocks, 128, 0, stream>>>(
            h1, W2 + (size_t)layer * DIM2 * DIM, b2 + (size_t)layer * DIM,
            scale, shift, xout);

        xin = xout;
    }
}